// kmeans_distribution_layer_88562225643942
// MI455X (gfx1250) — compile-verified
//
#include <hip/hip_runtime.h>
#include <math.h>

#define BATCH 8192
#define DIM   128
#define NCLS  128

typedef __attribute__((ext_vector_type(2))) float v2f;
typedef __attribute__((ext_vector_type(8))) float v8f;

// ---- workspace layout (float offsets) ----
// count    : [0, 128)
// sum_x    : [128, 128+16384)
// emp_mean : [16512, 32896)
// assign   : [32896, 41088) as int
#define WS_COUNT  0
#define WS_SUMX   128
#define WS_MEAN   (128 + DIM*NCLS)
#define WS_ASSIGN (128 + 2*DIM*NCLS)

// ---- K0: zero count + sum_x ----
__global__ void k0_zero(float* ws) {
    int i = blockIdx.x * blockDim.x + threadIdx.x;
    if (i < 128 + DIM*NCLS) ws[i] = 0.0f;
}

// ---- K1: WMMA distance GEMM + responsibilities + assignment stats ----
__global__ __launch_bounds__(256)
void k1_dist(const float* __restrict__ x, const float* __restrict__ centers,
             float* __restrict__ out, float* __restrict__ ws) {
    __shared__ float cn[NCLS];
    __shared__ float xn[16];
    __shared__ float dot_s[16][NCLS];
    __shared__ float pbuf[16][NCLS];
    __shared__ int   asg[16];

    const int tid  = threadIdx.x;
    const int row0 = blockIdx.x * 16;

    // norms
    if (tid < NCLS) {
        float s = 0.f;
        const float* c = centers + tid * DIM;
        for (int k = 0; k < DIM; ++k) s += c[k] * c[k];
        cn[tid] = s;
    }
    if (tid < 16) {
        float s = 0.f;
        const float* xr = x + (size_t)(row0 + tid) * DIM;
        for (int k = 0; k < DIM; ++k) s += xr[k] * xr[k];
        xn[tid] = s;
    }
    __syncthreads();

    // ---- WMMA phase: wave w computes columns [16w, 16w+16) ----
    const int wave = tid >> 5;
    const int lane = tid & 31;
    const int half = lane >> 4;   // 0: lanes 0-15, 1: lanes 16-31
    const int l16  = lane & 15;
    const int col0 = wave * 16;

    v8f acc = {};
    for (int k = 0; k < DIM; k += 4) {
        // A: 16x4 tile of x rows; lane holds K = 2*half + {0,1}
        v2f a = *(const v2f*)(x + (size_t)(row0 + l16) * DIM + k + 2 * half);
        // B: 4x16 tile of centers^T; lane holds column n=l16, same K pair
        v2f b = *(const v2f*)(centers + (size_t)(col0 + l16) * DIM + k + 2 * half);
        acc = __builtin_amdgcn_wmma_f32_16x16x4_f32(
            false, a, false, b, (short)0, acc, false, false);
    }
    // D layout: VGPR v -> row (v + 8*half), col l16
    for (int v = 0; v < 8; ++v) {
        int m = v + 8 * half;
        dot_s[m][col0 + l16] = acc[v];
    }
    __syncthreads();

    // ---- softmax / argmin phase: one worker thread per row ----
    if (tid < 16) {
        const int m = tid;
        float dsmin = 1e30f; int amin = 0;
        for (int j = 0; j < NCLS; ++j) {
            float ds = xn[m] + cn[j] - 2.f * dot_s[m][j];
            if (ds < dsmin) { dsmin = ds; amin = j; }
        }
        asg[m] = amin;
        float emax = -0.5f * dsmin;           // max of -0.5*ds
        float sumk = 0.f, pmax = -1e30f;
        for (int j = 0; j < NCLS; ++j) {
            float ds = xn[m] + cn[j] - 2.f * dot_s[m][j];
            sumk += __expf(-0.5f * ds - emax);
            float p = __expf(-64.5f * __logf(ds + 1.f));  // (ds+1)^(-(1+D)/2)
            pbuf[m][j] = p;
            pmax = fmaxf(pmax, p);
        }
        float sumc = 0.f;
        for (int j = 0; j < NCLS; ++j) sumc += __expf(pbuf[m][j] - pmax);
        const float LOGMIN = -18.420680743952367f;  // log(1e-8)
        float lk = __logf(sumk), lc = __logf(sumc);
        for (int j = 0; j < NCLS; ++j) {
            float ds = xn[m] + cn[j] - 2.f * dot_s[m][j];
            out[(size_t)(row0 + m) * NCLS + j] =
                fmaxf((-0.5f * ds - emax) - lk, LOGMIN);
            out[(size_t)BATCH * NCLS + (size_t)(row0 + m) * NCLS + j] =
                fmaxf((pbuf[m][j] - pmax) - lc, LOGMIN);
        }
        atomicAdd(ws + WS_COUNT + amin, 1.0f);
        ((int*)(ws + WS_ASSIGN))[row0 + m] = amin;
    }
    __syncthreads();

    // ---- scatter x rows into sum_x[assign] ----
    for (int idx = tid; idx < 16 * DIM; idx += 256) {
        int m = idx >> 7, d = idx & (DIM - 1);
        atomicAdd(ws + WS_SUMX + asg[m] * DIM + d,
                  x[(size_t)(row0 + m) * DIM + d]);
    }
}

// ---- K2: per-cluster means, cluster_dist, mean-MSE; zero covar-MSE slot ----
__global__ __launch_bounds__(128)
void k2_means(const float* __restrict__ centers,
              float* __restrict__ out, float* __restrict__ ws) {
    __shared__ float red[128];
    const int c = threadIdx.x;
    float n = ws[WS_COUNT + c];
    float invEx  = 1.f / fmaxf(n, 1.f);
    float invEps = 1.f / (n + 1e-7f);
    float sq = 0.f, mdsq = 0.f;
    for (int d = 0; d < DIM; ++d) {
        float s  = ws[WS_SUMX + c * DIM + d];
        float ce = centers[c * DIM + d];
        float me = s * invEx;           // exact mean
        float de = ce - me;
        sq += de * de;
        float em = s * invEps;          // eps-regularized mean
        ws[WS_MEAN + c * DIM + d] = em;
        float dm = em - ce;
        mdsq += dm * dm;
    }
    const size_t base = (size_t)2 * BATCH * NCLS;
    out[base + c] = (n > 0.f) ? sqrtf(sq) : 0.f;
    red[c] = n * mdsq;
    __syncthreads();
    for (int off = 64; off > 0; off >>= 1) {
        if (c < off) red[c] += red[c + off];
        __syncthreads();
    }
    if (c == 0) {
        out[base + NCLS]     = red[0] / (float)(BATCH * DIM);
        out[base + NCLS + 1] = 0.f;     // covar MSE accumulator
    }
}

// ---- K3: per-cluster covariance statistic, LDS-resident Gram matrix ----
__global__ __launch_bounds__(256)
void k3_covar(const float* __restrict__ x,
              float* __restrict__ out, const float* __restrict__ ws) {
    __shared__ float acc[DIM * DIM];     // 64 KB
    __shared__ float mvec[DIM];
    __shared__ float dbuf[8][DIM];
    __shared__ float red[256];

    const int c    = blockIdx.x;
    const int tid  = threadIdx.x;
    const int wave = tid >> 5;
    const int lane = tid & 31;

    for (int i = tid; i < DIM * DIM; i += 256) acc[i] = 0.f;
    if (tid < DIM) mvec[tid] = ws[WS_MEAN + c * DIM + tid];
    __syncthreads();

    const int* assign = (const int*)(ws + WS_ASSIGN);
    for (int b = wave; b < BATCH; b += 8) {        // wave-strided scan
        if (assign[b] != c) continue;
        const float* xr = x + (size_t)b * DIM;
        for (int r = 0; r < 4; ++r) {
            int i = lane + 32 * r;
            dbuf[wave][i] = xr[i] - mvec[i];
        }
        // intra-wave LDS RAW: drain DS counter before cross-lane reads
        asm volatile("s_wait_dscnt 0x0" ::: "memory");
        __builtin_amdgcn_wave_barrier();
        for (int r = 0; r < 4; ++r) {
            int i = lane + 32 * r;
            float di = dbuf[wave][i];
            for (int j = 0; j < DIM; ++j)
                atomicAdd(&acc[i * DIM + j], di * dbuf[wave][j]);  // ds_add_f32
        }
    }
    __syncthreads();

    float n   = ws[WS_COUNT + c];
    float inv = 1.f / (n + 1e-7f);
    float offp = 0.f;
    for (int idx = tid; idx < DIM * DIM; idx += 256) {
        int i = idx >> 7, j = idx & (DIM - 1);
        if (i != j) { float v = acc[idx] * inv; offp += v * v; }
    }
    float diagp = 0.f;
    if (tid < DIM) {
        float v = acc[tid * DIM + tid] * inv - 1.f;
        diagp = v * v;
    }
    red[tid] = n * diagp / (float)(BATCH * DIM)
             + n * offp  / ((float)BATCH * DIM * (DIM - 1));
    __syncthreads();
    for (int off = 128; off > 0; off >>= 1) {
        if (tid < off) red[tid] += red[tid + off];
        __syncthreads();
    }
    if (tid == 0)
        atomicAdd(&out[(size_t)2 * BATCH * NCLS + NCLS + 1], red[0]);
}

extern "C" void kernel_launch(void* const* d_in, const int* in_sizes, int n_in,
                              void* d_out, int out_size, void* d_ws, size_t ws_size,
                              hipStream_t stream) {
    const float* x       = (const float*)d_in[0];
    const float* centers = (const float*)d_in[1];
    float* out = (float*)d_out;
    float* ws  = (float*)d_ws;

    k0_zero <<<(128 + DIM*NCLS + 255) / 256, 256, 0, stream>>>(ws);
    k1_dist <<<BATCH / 16, 256, 0, stream>>>(x, centers, out, ws);
    k2_means<<<1, 128, 0, stream>>>(centers, out, ws);
    k3_covar<<<NCLS, 256, 0, stream>>>(x, out, ws);
}